// PolyDiffusionFold_16750372454443
// MI455X (gfx1250) — compile-verified
//
#include <hip/hip_runtime.h>

#ifndef __has_builtin
#define __has_builtin(x) 0
#endif

#define WAVE   32
#define CHUNK  8            // edges staged per wave (ASYNCcnt depth)
#define WPB    8            // waves per block
#define TPB    (WAVE * WPB) // 256 threads
#define DFEAT  128          // feature dim (reference constant)
#define GROUPS 8            // 16-element groups per wave in the WMMA combine

typedef int   v4i __attribute__((ext_vector_type(4)));
typedef float v2f __attribute__((ext_vector_type(2)));
typedef float v8f __attribute__((ext_vector_type(8)));
typedef __attribute__((address_space(1))) v4i* gas_v4i_ptr;
typedef __attribute__((address_space(3))) v4i* las_v4i_ptr;

// ---------------------------------------------------------------------------
// Wait for all outstanding async (LDS<-global) transfers of this wave.
// ---------------------------------------------------------------------------
__device__ __forceinline__ void wait_async_all() {
#if __has_builtin(__builtin_amdgcn_s_wait_asynccnt)
    __builtin_amdgcn_s_wait_asynccnt(0);
#else
    asm volatile("s_wait_asynccnt 0" ::: "memory");
#endif
}

// ---------------------------------------------------------------------------
// Tiny softmax over (K+1) logits -> alpha, single wave.
// ---------------------------------------------------------------------------
__global__ void softmax_small(const float* __restrict__ logits,
                              float* __restrict__ alpha, int n) {
    if (threadIdx.x == 0 && blockIdx.x == 0) {
        float m = logits[0];
        for (int i = 1; i < n; ++i) m = fmaxf(m, logits[i]);
        float s = 0.f;
        for (int i = 0; i < n; ++i) {
            float e = expf(logits[i] - m);
            alpha[i] = e;
            s += e;
        }
        float inv = 1.f / s;
        for (int i = 0; i < n; ++i) alpha[i] *= inv;
    }
}

// ---------------------------------------------------------------------------
// Fallback elementwise kernels (used only when K+1 != 4).
// ---------------------------------------------------------------------------
__global__ __launch_bounds__(256) void scale_vec(const float4* __restrict__ x,
                                                 float4* __restrict__ y,
                                                 const float* __restrict__ alpha,
                                                 long n4) {
    long i = (long)blockIdx.x * blockDim.x + threadIdx.x;
    if (i < n4) {
        const float a = alpha[0];
        float4 v = x[i];
        y[i] = make_float4(a * v.x, a * v.y, a * v.z, a * v.w);
    }
}

__global__ __launch_bounds__(256) void axpy_vec(float4* __restrict__ y,
                                                const float4* __restrict__ t,
                                                const float* __restrict__ alpha,
                                                int k, long n4) {
    long i = (long)blockIdx.x * blockDim.x + threadIdx.x;
    if (i < n4) {
        const float a = alpha[k];
        float4 v = t[i];
        float4 o = y[i];
        o.x += a * v.x; o.y += a * v.y; o.z += a * v.z; o.w += a * v.w;
        y[i] = o;
    }
}

// ---------------------------------------------------------------------------
// Fused polynomial combine via V_WMMA_F32_16X16X4_F32:
//   Y[p] = a0*t0[p] + a1*t1[p] + a2*t2[p] + a3*t3[p]
// One wave handles GROUPS groups of 16 consecutive elements. Per group:
//   A (16x4): every row = (a0,a1,a2,a3).
//     Layout: lane m<16 -> {v0=A[m][0], v1=A[m][1]}, lane m+16 -> {A[m][2], A[m][3]}
//   B (4x16): column n = (t0[p], t1[p], t2[p], t3[p]), p = gbase + n.
//     Layout: lane n<16 -> {v0=B[0][n], v1=B[1][n]}, lane n+16 -> {B[2][n], B[3][n]}
//   D (16x16): row 0 (v0, lanes 0-15) = combined outputs for the 16 elements.
// EXEC must be all-1s for WMMA, so the in-bounds test is wave-uniform; the
// ragged tail takes a scalar path.
// ---------------------------------------------------------------------------
__global__ __launch_bounds__(TPB) void combine_wmma(
        const float* __restrict__ t0, const float* __restrict__ t1,
        const float* __restrict__ t2, const float* __restrict__ t3,
        const float* __restrict__ alpha, float* __restrict__ Y, long n) {
    const int  lane = threadIdx.x & (WAVE - 1);
    const int  wib  = threadIdx.x / WAVE;
    const long wid  = (long)blockIdx.x * WPB + wib;
    const int  n16  = lane & 15;
    const bool hi   = lane >= 16;

    v2f a;
    a.x = hi ? alpha[2] : alpha[0];
    a.y = hi ? alpha[3] : alpha[1];

    const float* __restrict__ pX = hi ? t2 : t0;
    const float* __restrict__ pY = hi ? t3 : t1;

    const long base = wid * (16L * GROUPS);
    for (int g = 0; g < GROUPS; ++g) {
        const long gbase = base + (long)g * 16;
        if (gbase + 16 <= n) {             // wave-uniform: EXEC stays all-1s
            v2f b;
            b.x = pX[gbase + n16];
            b.y = pY[gbase + n16];
            v8f c = {};
            v8f d = __builtin_amdgcn_wmma_f32_16x16x4_f32(
                false, a, false, b, (short)0, c, false, false);
            if (!hi) Y[gbase + n16] = d[0];
        } else {                           // scalar ragged tail
            const long idx = gbase + n16;
            if (!hi && idx < n) {
                Y[idx] = alpha[0] * t0[idx] + alpha[1] * t1[idx] +
                         alpha[2] * t2[idx] + alpha[3] * t3[idx];
            }
        }
    }
}

// ---------------------------------------------------------------------------
// COO SpMM: for each edge e, Y[row[e], :] += val[e] * X[col[e], :]
// One wave per CHUNK edges. Each lane owns a 16B (float4) slice of the 512B
// feature row: a single GLOBAL_LOAD_ASYNC_TO_LDS_B128 gathers a full row into
// the wave's LDS stage. CHUNK gathers are batched in flight (ASYNCcnt), then
// consumed from LDS and scattered via global_atomic_add_f32 (no-return).
// ---------------------------------------------------------------------------
__global__ __launch_bounds__(TPB) void spmm_gather_scatter(
        const float* __restrict__ X, const int* __restrict__ erow,
        const int* __restrict__ ecol, const float* __restrict__ eval,
        float* __restrict__ Y, int nE) {
    __shared__ float stage[WPB * CHUNK * DFEAT]; // 32 KB / block

    const int  lane = threadIdx.x & (WAVE - 1);
    const int  wib  = threadIdx.x / WAVE;
    const long wid  = (long)blockIdx.x * WPB + wib;
    const long e0   = wid * CHUNK;
    if (e0 >= nE) return;

    const long rem = (long)nE - e0;
    const int  cnt = rem < CHUNK ? (int)rem : CHUNK;
    float* st = &stage[wib * (CHUNK * DFEAT)];

#if __has_builtin(__builtin_amdgcn_global_load_async_to_lds_b128)
    // Phase 1: issue per-lane-addressed async gathers (memory -> LDS).
    for (int i = 0; i < cnt; ++i) {
        const int    c   = ecol[e0 + i];
        const float* src = X + (size_t)c * DFEAT + lane * 4;
        float*       dst = st + i * DFEAT + lane * 4;
        __builtin_amdgcn_global_load_async_to_lds_b128(
            (gas_v4i_ptr)(void*)src,
            (las_v4i_ptr)(void*)dst,
            /*imm offset*/ 0, /*cpol*/ 0);
    }
    wait_async_all();
    // Phase 2: consume from LDS, scale, scatter-add.
    for (int i = 0; i < cnt; ++i) {
        const int    r  = erow[e0 + i];
        const float  v  = eval[e0 + i];
        const float4 xv = *(const float4*)(st + i * DFEAT + lane * 4);
        float* out = Y + (size_t)r * DFEAT + lane * 4;
        atomicAdd(out + 0, v * xv.x);
        atomicAdd(out + 1, v * xv.y);
        atomicAdd(out + 2, v * xv.z);
        atomicAdd(out + 3, v * xv.w);
    }
#else
    // Fallback: direct VGPR gather (still b128 vectorized) + atomic scatter.
    (void)st;
    for (int i = 0; i < cnt; ++i) {
        const int    c  = ecol[e0 + i];
        const int    r  = erow[e0 + i];
        const float  v  = eval[e0 + i];
        const float4 xv = *(const float4*)(X + (size_t)c * DFEAT + lane * 4);
        float* out = Y + (size_t)r * DFEAT + lane * 4;
        atomicAdd(out + 0, v * xv.x);
        atomicAdd(out + 1, v * xv.y);
        atomicAdd(out + 2, v * xv.z);
        atomicAdd(out + 3, v * xv.w);
    }
#endif
}

// ---------------------------------------------------------------------------
// Host-side orchestration (graph-capture safe: only async ops on `stream`).
// Workspace layout: [alpha: 256 B][buf1][buf2][buf3], each N*D f32.
// ---------------------------------------------------------------------------
extern "C" void kernel_launch(void* const* d_in, const int* in_sizes, int n_in,
                              void* d_out, int out_size, void* d_ws, size_t ws_size,
                              hipStream_t stream) {
    const float* x      = (const float*)d_in[0];
    const int*   erow   = (const int*)d_in[1];
    const int*   ecol   = (const int*)d_in[2];
    const float* evals  = (const float*)d_in[3];
    const float* logits = (const float*)d_in[4];

    const int  ND  = in_sizes[0];   // N * 128
    const int  E   = in_sizes[1];
    const int  Kp1 = in_sizes[4];   // K + 1
    const long n4  = (long)ND / 4;

    float* Y     = (float*)d_out;
    float* alpha = (float*)d_ws;
    float* buf1  = (float*)((char*)d_ws + 256);
    float* buf2  = buf1 + ND;
    float* buf3  = buf2 + ND;

    softmax_small<<<1, 32, 0, stream>>>(logits, alpha, Kp1);

    const int bSpmm = (E + CHUNK * WPB - 1) / (CHUNK * WPB);

    if (Kp1 == 4) {
        // Compute the three diffusion terms, then one fused WMMA combine.
        float* bufs[3] = { buf1, buf2, buf3 };
        const float* src = x;
        for (int k = 0; k < 3; ++k) {
            float* dst = bufs[k];
            (void)hipMemsetAsync(dst, 0, (size_t)ND * sizeof(float), stream);
            spmm_gather_scatter<<<bSpmm, TPB, 0, stream>>>(src, erow, ecol, evals, dst, E);
            src = dst;
        }
        const long groups = ((long)ND + 15) / 16;
        const long waves  = (groups + GROUPS - 1) / GROUPS;
        const int  bComb  = (int)((waves + WPB - 1) / WPB);
        combine_wmma<<<bComb, TPB, 0, stream>>>(x, buf1, buf2, buf3, alpha, Y, (long)ND);
    } else {
        // Generic K: incremental scale/axpy chain with ping-pong buffers.
        const int bElem = (int)((n4 + 255) / 256);
        scale_vec<<<bElem, 256, 0, stream>>>((const float4*)x, (float4*)Y, alpha, n4);
        const float* src = x;
        float* bufs[2] = { buf1, buf2 };
        for (int k = 1; k < Kp1; ++k) {
            float* dst = bufs[k & 1];
            (void)hipMemsetAsync(dst, 0, (size_t)ND * sizeof(float), stream);
            spmm_gather_scatter<<<bSpmm, TPB, 0, stream>>>(src, erow, ecol, evals, dst, E);
            axpy_vec<<<bElem, 256, 0, stream>>>((float4*)Y, (const float4*)dst, alpha, k, n4);
            src = dst;
        }
    }
}